// AttnDecoderRNN_69114613729281
// MI455X (gfx1250) — compile-verified
//
#include <hip/hip_runtime.h>
#include <hip/hip_bf16.h>
#include <math.h>

#define H    1024
#define H2   2048
#define H3   3072
#define VV   50257
#define SS   8192

typedef __attribute__((ext_vector_type(2))) float v2f;
typedef __attribute__((ext_vector_type(8))) float v8f;

// ---------------- reductions (wave32) ----------------
__device__ __forceinline__ float wave_sum(float v) {
#pragma unroll
    for (int off = 16; off > 0; off >>= 1) v += __shfl_xor(v, off, 32);
    return v;
}
__device__ __forceinline__ float wave_max(float v) {
#pragma unroll
    for (int off = 16; off > 0; off >>= 1) v = fmaxf(v, __shfl_xor(v, off, 32));
    return v;
}
__device__ __forceinline__ float block_sum(float v, float* sm) {
    int lane = threadIdx.x & 31, wid = threadIdx.x >> 5;
    v = wave_sum(v);
    if (lane == 0) sm[wid] = v;
    __syncthreads();
    int nw = (blockDim.x + 31) >> 5;
    if (wid == 0) {
        float r = (lane < nw) ? sm[lane] : 0.0f;
        r = wave_sum(r);
        if (lane == 0) sm[0] = r;
    }
    __syncthreads();
    float r = sm[0];
    __syncthreads();
    return r;
}
__device__ __forceinline__ float block_max(float v, float* sm) {
    int lane = threadIdx.x & 31, wid = threadIdx.x >> 5;
    v = wave_max(v);
    if (lane == 0) sm[wid] = v;
    __syncthreads();
    int nw = (blockDim.x + 31) >> 5;
    if (wid == 0) {
        float r = (lane < nw) ? sm[lane] : -INFINITY;
        r = wave_max(r);
        if (lane == 0) sm[0] = r;
    }
    __syncthreads();
    float r = sm[0];
    __syncthreads();
    return r;
}

// ---------------- K1: x = [emb[word], last_context] ----------------
__global__ void k_build_x(const int* word, const float* emb, const float* last_ctx, float* x) {
    int i = blockIdx.x * 256 + threadIdx.x;           // 2048 threads
    int w = word[0];
    x[i] = (i < H) ? emb[(size_t)w * H + i] : last_ctx[i - H];
}

// ---------------- K2: gi = W_ih x + b_ih ; gh = W_hh h + b_hh (wave per row) ----------------
__global__ void k_gru_gemv(const float* __restrict__ W_ih, const float* __restrict__ W_hh,
                           const float* __restrict__ b_ih, const float* __restrict__ b_hh,
                           const float* __restrict__ x, const float* __restrict__ h,
                           float* gi, float* gh) {
    int w = blockIdx.x * 8 + (threadIdx.x >> 5);      // 0..6143
    int lane = threadIdx.x & 31;
    float acc = 0.0f;
    if (w < H3) {
        const float4* Wr = (const float4*)(W_ih + (size_t)w * H2);
        const float4* xr = (const float4*)x;
#pragma unroll
        for (int k = 0; k < 16; ++k) {
            float4 a = Wr[lane + k * 32];
            float4 b = xr[lane + k * 32];
            acc += a.x * b.x + a.y * b.y + a.z * b.z + a.w * b.w;
        }
        acc = wave_sum(acc);
        if (lane == 0) gi[w] = acc + b_ih[w];
    } else {
        int r = w - H3;
        const float4* Wr = (const float4*)(W_hh + (size_t)r * H);
        const float4* hr = (const float4*)h;
#pragma unroll
        for (int k = 0; k < 8; ++k) {
            float4 a = Wr[lane + k * 32];
            float4 b = hr[lane + k * 32];
            acc += a.x * b.x + a.y * b.y + a.z * b.z + a.w * b.w;
        }
        acc = wave_sum(acc);
        if (lane == 0) gh[r] = acc + b_hh[r];
    }
}

// ---------------- K3: GRU gate fusion ----------------
__global__ void k_gru_combine(const float* gi, const float* gh, const float* hprev,
                              float* h_new, float* ybuf, float* out_hidden) {
    int i = blockIdx.x * 256 + threadIdx.x;           // 1024 threads
    float r = 1.0f / (1.0f + expf(-(gi[i] + gh[i])));
    float z = 1.0f / (1.0f + expf(-(gi[H + i] + gh[H + i])));
    float n = tanhf(gi[2 * H + i] + r * gh[2 * H + i]);
    float hn = (1.0f - z) * n + z * hprev[i];
    h_new[i] = hn;
    ybuf[i] = hn;
    out_hidden[i] = hn;
}

// ---------------- K4: v = attn_W^T h  (thread per output column) ----------------
__global__ void k_attn_v(const float* __restrict__ attn_W, const float* __restrict__ h, float* v) {
    __shared__ float hs[H];
    for (int i = threadIdx.x; i < H; i += 256) hs[i] = h[i];
    __syncthreads();
    int j = blockIdx.x * 256 + threadIdx.x;           // 1024 threads
    float acc = 0.0f;
#pragma unroll 4
    for (int i = 0; i < H; ++i) acc += hs[i] * attn_W[(size_t)i * H + j];
    v[j] = acc;
}

// ---------------- K4c: c = attn_b . h ----------------
__global__ void k_attn_c(const float* attn_b, const float* h, float* cOut) {
    __shared__ float sm[32];
    float acc = 0.0f;
    for (int i = threadIdx.x; i < H; i += 256) acc += attn_b[i] * h[i];
    float s = block_sum(acc, sm);
    if (threadIdx.x == 0) cOut[0] = s;
}

// ---------------- K5: scores[s] = enc[s,:] . v + c (wave per row) ----------------
__global__ void k_scores(const float* __restrict__ enc, const float* __restrict__ v,
                         const float* cptr, float* scores) {
    int s = blockIdx.x * 8 + (threadIdx.x >> 5);      // 0..8191
    int lane = threadIdx.x & 31;
    const float4* er = (const float4*)(enc + (size_t)s * H);
    const float4* vr = (const float4*)v;
    float acc = 0.0f;
#pragma unroll
    for (int k = 0; k < 8; ++k) {
        float4 a = er[lane + k * 32];
        float4 b = vr[lane + k * 32];
        acc += a.x * b.x + a.y * b.y + a.z * b.z + a.w * b.w;
    }
    acc = wave_sum(acc);
    if (lane == 0) scores[s] = acc + cptr[0];
}

// ---------------- K6: softmax over S=8192 (1 block x 1024) ----------------
__global__ void k_softmax(const float* scores, float* weights, float* out_attn) {
    __shared__ float sm[32];
    float e[8];
    float lmax = -INFINITY;
#pragma unroll
    for (int k = 0; k < 8; ++k) lmax = fmaxf(lmax, scores[threadIdx.x + k * 1024]);
    float M = block_max(lmax, sm);
    float lsum = 0.0f;
#pragma unroll
    for (int k = 0; k < 8; ++k) { e[k] = expf(scores[threadIdx.x + k * 1024] - M); lsum += e[k]; }
    float Z = block_sum(lsum, sm);
    float inv = 1.0f / Z;
#pragma unroll
    for (int k = 0; k < 8; ++k) {
        int s = threadIdx.x + k * 1024;
        float wv = e[k] * inv;
        weights[s] = wv;
        out_attn[s] = wv;
    }
}

// ---------------- K7: context[j] = sum_s w[s]*enc[s,j] ----------------
__global__ void k_context(const float* __restrict__ weights, const float* __restrict__ enc,
                          float* ybuf_ctx, float* out_ctx) {
    int j = blockIdx.x * 256 + threadIdx.x;           // 1024 threads
    float acc = 0.0f;
#pragma unroll 4
    for (int s = 0; s < SS; ++s) acc += weights[s] * enc[(size_t)s * H + j];
    ybuf_ctx[j] = acc;
    out_ctx[j] = acc;
}

// ---------------- K8: logits = out_W @ [h_new, context] + out_b via WMMA f32 16x16x4 ----------------
// Each wave owns a 16-row tile of out_W; A = 16x4 f32 tile (ISA A-layout: lanes 0-15 K=0,1;
// lanes 16-31 K=2,3), B = 4-elem slice of y broadcast to all 16 columns (from LDS).
// Two independent accumulator chains (even/odd K-chunks) double intra-wave WMMA ILP;
// since every D column holds the same dot-product partials, c0+c1 is the exact result.
// D column 0: rows 0-7 @ lane0 VGPR0-7, rows 8-15 @ lane16 VGPR0-7.
__global__ void k_out_gemv_wmma(const float* __restrict__ out_W, const float* __restrict__ out_b,
                                const float* __restrict__ y, float* logits) {
    __shared__ float ys[H2];
    for (int i = threadIdx.x; i < H2 / 4; i += 256)
        ((float4*)ys)[i] = ((const float4*)y)[i];
    __syncthreads();

    int wave = threadIdx.x >> 5;
    int lane = threadIdx.x & 31;
    int base = blockIdx.x * 128 + wave * 16;
    int m = lane & 15;
    int koff = (lane >> 4) << 1;                      // 0 for lanes 0-15, 2 for lanes 16-31
    int rowIdx = base + m;
    int rc = rowIdx < VV ? rowIdx : (VV - 1);         // clamp loads; stores are guarded
    const float* Wr = out_W + (size_t)rc * H2 + koff;
    const float* yb = ys + koff;

    v8f c0 = {};
    v8f c1 = {};
#pragma unroll 4
    for (int kk = 0; kk < H2 / 4; kk += 2) {
        v2f a0 = *(const v2f*)(Wr + kk * 4);          // global_load_b64, 8B aligned
        v2f b0 = *(const v2f*)(yb + kk * 4);          // ds_load_b64 (broadcast within half-wave)
        v2f a1 = *(const v2f*)(Wr + kk * 4 + 4);
        v2f b1 = *(const v2f*)(yb + kk * 4 + 4);
        c0 = __builtin_amdgcn_wmma_f32_16x16x4_f32(
                false, a0, false, b0, (short)0, c0, false, false);
        c1 = __builtin_amdgcn_wmma_f32_16x16x4_f32(
                false, a1, false, b1, (short)0, c1, false, false);
    }

    if (lane == 0 || lane == 16) {
        int rbase = base + ((lane == 16) ? 8 : 0);
#pragma unroll
        for (int j = 0; j < 8; ++j) {
            int r = rbase + j;
            if (r < VV) logits[r] = (c0[j] + c1[j]) + out_b[r];
        }
    }
}

// ---------------- K9: log_softmax over V (1 block x 1024) ----------------
__global__ void k_logsoftmax(const float* __restrict__ logits, float* __restrict__ outp) {
    __shared__ float sm[32];
    float lmax = -INFINITY;
    for (int i = threadIdx.x; i < VV; i += 1024) lmax = fmaxf(lmax, logits[i]);
    float M = block_max(lmax, sm);
    float lsum = 0.0f;
    for (int i = threadIdx.x; i < VV; i += 1024) lsum += expf(logits[i] - M);
    float Z = block_sum(lsum, sm);
    float lse = M + logf(Z);
    for (int i = threadIdx.x; i < VV; i += 1024) outp[i] = logits[i] - lse;
}

// ---------------- launch ----------------
extern "C" void kernel_launch(void* const* d_in, const int* in_sizes, int n_in,
                              void* d_out, int out_size, void* d_ws, size_t ws_size,
                              hipStream_t stream) {
    const int*   word     = (const int*)d_in[0];
    const float* last_ctx = (const float*)d_in[1];
    const float* last_hid = (const float*)d_in[2];
    const float* enc      = (const float*)d_in[3];
    const float* emb      = (const float*)d_in[4];
    const float* W_ih     = (const float*)d_in[5];
    const float* W_hh     = (const float*)d_in[6];
    const float* b_ih     = (const float*)d_in[7];
    const float* b_hh     = (const float*)d_in[8];
    const float* attn_W   = (const float*)d_in[9];
    const float* attn_b   = (const float*)d_in[10];
    const float* out_W    = (const float*)d_in[11];
    const float* out_b    = (const float*)d_in[12];

    float* out = (float*)d_out;
    float* out_log  = out;                 // 50257
    float* out_ctx  = out + VV;            // 1024
    float* out_hid  = out + VV + H;        // 1024
    float* out_attn = out + VV + 2 * H;    // 8192

    float* ws = (float*)d_ws;
    float* x      = ws + 0;        // 2048
    float* gi     = ws + 2048;     // 3072
    float* gh     = ws + 5120;     // 3072
    float* h_new  = ws + 8192;     // 1024
    float* v      = ws + 9216;     // 1024
    float* cval   = ws + 10240;    // 1
    float* scores = ws + 10496;    // 8192
    float* wts    = ws + 18688;    // 8192
    float* ybuf   = ws + 26880;    // 2048 = [h_new, context]
    float* logits = ws + 28928;    // 50257

    k_build_x<<<8, 256, 0, stream>>>(word, emb, last_ctx, x);
    k_gru_gemv<<<768, 256, 0, stream>>>(W_ih, W_hh, b_ih, b_hh, x, last_hid, gi, gh);
    k_gru_combine<<<4, 256, 0, stream>>>(gi, gh, last_hid, h_new, ybuf, out_hid);
    k_attn_v<<<4, 256, 0, stream>>>(attn_W, h_new, v);
    k_attn_c<<<1, 256, 0, stream>>>(attn_b, h_new, cval);
    k_scores<<<1024, 256, 0, stream>>>(enc, v, cval, scores);
    k_softmax<<<1, 1024, 0, stream>>>(scores, wts, out_attn);
    k_context<<<4, 256, 0, stream>>>(wts, enc, ybuf + H, out_ctx);
    k_out_gemv_wmma<<<(VV + 127) / 128, 256, 0, stream>>>(out_W, out_b, ybuf, logits);
    k_logsoftmax<<<1, 1024, 0, stream>>>(logits, out_log);
}